// MultiHeadPointTransformer_89086211653966
// MI455X (gfx1250) — compile-verified
//
#include <hip/hip_runtime.h>
#include <hip/hip_bf16.h>

typedef float v2f __attribute__((ext_vector_type(2)));
typedef float v8f __attribute__((ext_vector_type(8)));

#define IN_C 64
#define OUT_C 64
#define HD 16

// ---------------------------------------------------------------------------
// Init: node_max = -inf, den = 0, accum = 0
// ---------------------------------------------------------------------------
__global__ void pt_init_kernel(float* __restrict__ nmax, float* __restrict__ den,
                               float* __restrict__ accum, int tot) {
    int i = blockIdx.x * blockDim.x + threadIdx.x;
    if (i >= tot) return;
    nmax[i]  = -INFINITY;
    den[i]   = 0.0f;
    accum[i] = 0.0f;
}

// ---------------------------------------------------------------------------
// GEMM: Y[n x 64] = X[n x 64] @ W (+bias) (+relu)   via V_WMMA_F32_16X16X4_F32
// mode 0: W laid out [H=4][64][16]  (column = h*16 + d)
// mode 1: W laid out [64][64] row-major
// One wave32 computes one 16x16 output tile; K-loop = 16 WMMA ops.
// ---------------------------------------------------------------------------
__global__ __launch_bounds__(256)
void pt_gemm64_wmma(const float* __restrict__ X, const float* __restrict__ W,
                    const float* __restrict__ bias, float* __restrict__ Y,
                    int n, int mode, int relu) {
    const int wave  = threadIdx.x >> 5;
    const int lane  = threadIdx.x & 31;
    const int tile  = blockIdx.x * (blockDim.x >> 5) + wave;
    const int ntile = (n >> 4) * 4;              // n is a multiple of 16
    if (tile >= ntile) return;

    const int m0   = (tile >> 2) << 4;           // node-row tile base
    const int n0   = (tile & 3) << 4;            // output-column tile base
    const int half = lane >> 4;                  // 0: lanes 0-15, 1: lanes 16-31
    const int l15  = lane & 15;
    const int mrow = m0 + l15;                   // A-matrix row for this lane
    const int col  = n0 + l15;                   // B/C/D column for this lane
    const int hh   = col >> 4;                   // head (mode 0)
    const int dd   = col & 15;                   // within-head channel (mode 0)

    v8f acc = {};
    #pragma unroll
    for (int k = 0; k < 64; k += 4) {
        const int kk = k + half * 2;             // lanes 16-31 carry K+2, K+3
        // A 16x4 fp32: lane holds row mrow, VGPR0=K(kk), VGPR1=K(kk+1)
        v2f a = *(const v2f*)(X + mrow * 64 + kk);
        // B 4x16 fp32: VGPR0 = row K(kk) striped over lanes, VGPR1 = row K(kk+1)
        v2f b;
        if (mode == 0) {
            b.x = W[(hh * 64 + kk    ) * 16 + dd];
            b.y = W[(hh * 64 + kk + 1) * 16 + dd];
        } else {
            b.x = W[(kk    ) * 64 + col];
            b.y = W[(kk + 1) * 64 + col];
        }
        acc = __builtin_amdgcn_wmma_f32_16x16x4_f32(
            /*neg_a=*/false, a, /*neg_b=*/false, b,
            /*c_mod=*/(short)0, acc, /*reuse_a=*/false, /*reuse_b=*/false);
    }

    const float bv = bias ? bias[col] : 0.0f;
    #pragma unroll
    for (int r = 0; r < 8; ++r) {
        // D layout: VGPR r -> row m0+r (lanes 0-15) / m0+8+r (lanes 16-31)
        float v = acc[r] + bv;
        if (relu) v = fmaxf(v, 0.0f);
        Y[(m0 + r + half * 8) * 64 + col] = v;
    }
}

// ---------------------------------------------------------------------------
// Float atomic max via sign-aware int/uint atomics (native global_atomic_max/min)
// ---------------------------------------------------------------------------
__device__ __forceinline__ void atomicMaxF(float* addr, float val) {
    const int iv = __float_as_int(val);
    if (val >= 0.0f) atomicMax((int*)addr, iv);
    else             atomicMin((unsigned int*)addr, (unsigned int)iv);
}

// ---------------------------------------------------------------------------
// Edge passes: one thread per (edge, channel). 64 channels per edge.
//   pass 0: segment max of logits -> nmax
//   pass 1: segment sum of exp(logit - max) -> den
//   pass 2: scatter-add alpha * (x_lin[src] + delta) -> accum
// ---------------------------------------------------------------------------
__global__ __launch_bounds__(256)
void pt_edge_pass(const int* __restrict__ ei, const float* __restrict__ pos,
                  const float* __restrict__ Wpos, const float* __restrict__ bpos,
                  const float* __restrict__ asrc, const float* __restrict__ adst,
                  const float* __restrict__ xlin,
                  float* __restrict__ nmax, float* __restrict__ den,
                  float* __restrict__ accum, int E, int pass) {
    int t = blockIdx.x * blockDim.x + threadIdx.x;   // E*64 = 204.8M < 2^31
    if (t >= E * 64) return;
    const int e  = t >> 6;
    const int c  = t & 63;
    const int s  = ei[e];
    const int dn = ei[E + e];
    const int hh = c >> 4;
    const int dd = c & 15;

    const float rx = pos[s * 3 + 0] - pos[dn * 3 + 0];
    const float ry = pos[s * 3 + 1] - pos[dn * 3 + 1];
    const float rz = pos[s * 3 + 2] - pos[dn * 3 + 2];
    // delta = rel . W_pos[h,:,dd] + b_pos[h,dd]   (W_pos is [H,3,16])
    const float delta = bpos[c]
                      + rx * Wpos[(hh * 3 + 0) * 16 + dd]
                      + ry * Wpos[(hh * 3 + 1) * 16 + dd]
                      + rz * Wpos[(hh * 3 + 2) * 16 + dd];

    const float logit = asrc[s * 64 + c] - adst[dn * 64 + c] + delta;
    const int gi = dn * 64 + c;

    if (pass == 0) {
        atomicMaxF(&nmax[gi], logit);
    } else {
        const float ex = __expf(logit - nmax[gi]);
        if (pass == 1) {
            atomicAdd(&den[gi], ex);
        } else {
            const float alpha = ex / (den[gi] + 1e-16f);
            atomicAdd(&accum[gi], alpha * (xlin[s * 64 + c] + delta));
        }
    }
}

// ---------------------------------------------------------------------------
// Launch
// ---------------------------------------------------------------------------
extern "C" void kernel_launch(void* const* d_in, const int* in_sizes, int n_in,
                              void* d_out, int out_size, void* d_ws, size_t ws_size,
                              hipStream_t stream) {
    const float* x     = (const float*)d_in[0];
    const float* pos   = (const float*)d_in[1];
    const int*   ei    = (const int*)  d_in[2];
    const float* W_lin = (const float*)d_in[3];
    const float* W_src = (const float*)d_in[4];
    const float* W_dst = (const float*)d_in[5];
    const float* W_pos = (const float*)d_in[6];
    const float* b_pos = (const float*)d_in[7];
    const float* W1    = (const float*)d_in[8];
    const float* b1    = (const float*)d_in[9];
    const float* W2    = (const float*)d_in[10];
    const float* b2    = (const float*)d_in[11];
    float* out = (float*)d_out;

    const int n = in_sizes[0] / IN_C;        // 100000
    const int E = in_sizes[2] / 2;           // 3200000

    float* ws = (float*)d_ws;
    const size_t per = (size_t)n * 64;
    float* xlin  = ws + 0 * per;
    float* asrc  = ws + 1 * per;
    float* adst  = ws + 2 * per;
    float* nmax  = ws + 3 * per;
    float* den   = ws + 4 * per;
    float* accum = ws + 5 * per;
    float* hbuf  = nmax;                     // reuse after edge passes finish

    // init accumulators
    const int tot = n * 64;
    pt_init_kernel<<<(tot + 255) / 256, 256, 0, stream>>>(nmax, den, accum, tot);

    // node projections: x_lin, a_src, a_dst (head-blocked weight layout)
    const int gtiles  = (n / 16) * 4;
    const int gblocks = (gtiles + 7) / 8;    // 8 waves (one tile each) per block
    pt_gemm64_wmma<<<gblocks, 256, 0, stream>>>(x, W_lin, nullptr, xlin, n, 0, 0);
    pt_gemm64_wmma<<<gblocks, 256, 0, stream>>>(x, W_src, nullptr, asrc, n, 0, 0);
    pt_gemm64_wmma<<<gblocks, 256, 0, stream>>>(x, W_dst, nullptr, adst, n, 0, 0);

    // edge stage: 3 streaming passes (max, sum-exp, weighted scatter)
    const long long etot = (long long)E * 64;
    const int eblocks = (int)((etot + 255) / 256);
    pt_edge_pass<<<eblocks, 256, 0, stream>>>(ei, pos, W_pos, b_pos, asrc, adst,
                                              xlin, nmax, den, accum, E, 0);
    pt_edge_pass<<<eblocks, 256, 0, stream>>>(ei, pos, W_pos, b_pos, asrc, adst,
                                              xlin, nmax, den, accum, E, 1);
    pt_edge_pass<<<eblocks, 256, 0, stream>>>(ei, pos, W_pos, b_pos, asrc, adst,
                                              xlin, nmax, den, accum, E, 2);

    // projection MLP: relu(accum @ W1 + b1) @ W2 + b2  (plain weight layout)
    pt_gemm64_wmma<<<gblocks, 256, 0, stream>>>(accum, W1, b1, hbuf, n, 1, 1);
    pt_gemm64_wmma<<<gblocks, 256, 0, stream>>>(hbuf, W2, b2, out, n, 1, 0);
}